// Quantizer_38053410243327
// MI455X (gfx1250) — compile-verified
//
#include <hip/hip_runtime.h>
#include <hip/hip_fp16.h>

// Row layout: 64 outlier floats + 4096 core floats = 4160 floats per row.
#define POD_RANK   64
#define HIDDEN     4096
#define ROW_W      (POD_RANK + HIDDEN)
#define Q_MAX      7.0f
#define BLOCK      256          // 8 wave32 waves
#define F4_PER_THD 4            // 4 x float4 = 16 core floats per thread

typedef float v4f __attribute__((ext_vector_type(4)));

__global__ __launch_bounds__(BLOCK)
void quantizer_kernel(const float* __restrict__ x,
                      float* __restrict__ out,
                      int nrows)
{
    // Single shared object -> placed at LDS byte offset 0, which the raw
    // LDS offsets fed to global_load_async_to_lds_b128 below rely on.
    __shared__ __align__(16) float smem[HIDDEN + 8];

    const int tid = threadIdx.x;
    const int row = blockIdx.x;
    if (row >= nrows) return;

    const float* rowbase = x + (size_t)row * ROW_W;
    const float* core    = rowbase + POD_RANK;   // uniform per block -> SGPR pair

    // ---- Async DMA: 16 KB core row -> LDS (ASYNCcnt path, bypasses VGPRs) ----
    // GVS addressing: mem_addr = SADDR(64b) + VADDR(32b byte offset)
    // LDS destination: VDST holds the 32b LDS byte offset
    // Input is streamed exactly once (273 MB > 192 MB L2) -> non-temporal hint.
#pragma unroll
    for (int j = 0; j < F4_PER_THD; ++j) {
        unsigned byte_off = (unsigned)((tid + j * BLOCK) * 16);
        asm volatile("global_load_async_to_lds_b128 %0, %1, %2 th:TH_LOAD_NT"
                     :
                     : "v"(byte_off), "v"(byte_off), "s"(core)
                     : "memory");
    }
    asm volatile("s_wait_asynccnt 0" ::: "memory");

    // Each thread reads back exactly the bytes it DMA'd (no cross-thread
    // dependency yet, so no barrier needed before this pass).
    v4f r[F4_PER_THD];
    float amax = 0.0f;
#pragma unroll
    for (int j = 0; j < F4_PER_THD; ++j) {
        r[j] = *(const v4f*)&smem[(tid + j * BLOCK) * 4];
        amax = fmaxf(amax, fmaxf(fmaxf(fabsf(r[j].x), fabsf(r[j].y)),
                                 fmaxf(fabsf(r[j].z), fabsf(r[j].w))));
    }

    // ---- wave32 butterfly max ----
#pragma unroll
    for (int m = 16; m > 0; m >>= 1)
        amax = fmaxf(amax, __shfl_xor(amax, m, 32));

    const int wave = tid >> 5;
    if ((tid & 31) == 0) smem[HIDDEN + wave] = amax;
    __syncthreads();

    float rmax = smem[HIDDEN + 0];
#pragma unroll
    for (int w = 1; w < BLOCK / 32; ++w)
        rmax = fmaxf(rmax, smem[HIDDEN + w]);

    // scale = fp16(absmax/7) * fp16(1.0); quantize against its f32 value.
    const __half hs = __float2half(rmax * (1.0f / Q_MAX));
    const float  s  = __half2float(hs);
    const float  inv = 1.0f / s;   // one divide per thread, mul per element

    // ---- quantize + non-temporal b128 stores (output streamed once) ----
    v4f* outq = (v4f*)out;         // region 0: nrows*4096 quantized values
#pragma unroll
    for (int j = 0; j < F4_PER_THD; ++j) {
        v4f v = r[j], o;
        o.x = fminf(Q_MAX, fmaxf(-Q_MAX, rintf(v.x * inv)));
        o.y = fminf(Q_MAX, fmaxf(-Q_MAX, rintf(v.y * inv)));
        o.z = fminf(Q_MAX, fmaxf(-Q_MAX, rintf(v.z * inv)));
        o.w = fminf(Q_MAX, fmaxf(-Q_MAX, rintf(v.w * inv)));
        __builtin_nontemporal_store(o, &outq[(size_t)row * (HIDDEN / 4) + tid + j * BLOCK]);
    }

    const size_t q_elems = (size_t)nrows * HIDDEN;
    if (tid == 0)
        out[q_elems + row] = s;                    // region 1: scales (as f32)

    if (tid < POD_RANK / 4) {                      // region 2: outlier passthrough
        const v4f* op = (const v4f*)rowbase;
        v4f ov = __builtin_nontemporal_load(&op[tid]);
        v4f* oo = (v4f*)(out + q_elems + (size_t)nrows);
        __builtin_nontemporal_store(ov, &oo[(size_t)row * (POD_RANK / 4) + tid]);
    }
}

extern "C" void kernel_launch(void* const* d_in, const int* in_sizes, int n_in,
                              void* d_out, int out_size, void* d_ws, size_t ws_size,
                              hipStream_t stream)
{
    const float* x = (const float*)d_in[0];
    float* out = (float*)d_out;
    const int nrows = in_sizes[0] / ROW_W;   // 4*4096 = 16384
    quantizer_kernel<<<nrows, BLOCK, 0, stream>>>(x, out, nrows);
}